// yoloHeadv3_35150012350770
// MI455X (gfx1250) — compile-verified
//
#include <hip/hip_runtime.h>
#include <cstdint>

// ---------------------------------------------------------------------------
// YOLOv3 head decode, gfx1250 (MI455X).
// Input : X  (B, 255, 52, 52) f32   [255 = 3 anchors * 85 channels]
// Output: out(B, 3*52*52, 85) f32
// Per (b,a) slab: 85 x 2704 transpose + pointwise decode on channels 0..4.
// Memory bound: ~353 MB total -> ~15 us floor at 23.3 TB/s.
//
// Data path:
//   global --(per-lane async DMA, transpose scatter)--> LDS, position-major
//   LDS --(linear b128 reads)--> decode --> NT b128 coalesced stores.
// CDNA5-specific: GLOBAL_LOAD_ASYNC_TO_LDS_B32 (ASYNCcnt) + s_wait_asynccnt.
// ---------------------------------------------------------------------------

typedef float v4f __attribute__((ext_vector_type(4)));

namespace {
constexpr int kA       = 3;
constexpr int kG       = 52;
constexpr int kGG      = kG * kG;                 // 2704 positions per slab
constexpr int kCH      = 85;                      // 5 + 80 classes
constexpr int kPT      = 64;                      // positions per tile
constexpr int kTiles   = (kGG + kPT - 1) / kPT;   // 43 (last tile: 16 pos)
constexpr int kThreads = 256;
constexpr float kRatio    = 8.0f;                 // 416 / 52
constexpr float kExpClamp = 1000.0f;
}

// CDNA5 async DMA: copies the float at gptr to the LDS address in lds_off, per lane, tracked by ASYNCcnt.
__device__ __forceinline__ void async_f32_to_lds(unsigned lds_off, const float* gptr) {
  asm volatile("global_load_async_to_lds_b32 %0, %1, off"
               :: "v"(lds_off), "v"((unsigned long long)(uintptr_t)gptr)
               : "memory");
}

__device__ __forceinline__ float sigmoidf_(float v) {
  return 1.0f / (1.0f + expf(-v));
}

__global__ __launch_bounds__(kThreads) void yolo_head_kernel(
    const float* __restrict__ X, float* __restrict__ out) {
  __shared__ float tile[kPT * kCH];               // tile[p][c], 21.76 KB

  const int tid = threadIdx.x;
  const int blk = blockIdx.x;                     // b*kA*kTiles + a*kTiles + t
  const int t   = blk % kTiles;
  const int ba  = blk / kTiles;
  const int a   = ba % kA;
  const int b   = ba / kA;

  const int pos0 = t * kPT;
  const int npos = min(kPT, kGG - pos0);

  const float* slab  = X   + ((size_t)b * (kA * kCH) + (size_t)a * kCH) * kGG;
  float*       oslab = out + ((size_t)b * (kA * kGG) + (size_t)a * kGG) * kCH;

  // ---- Stage global -> LDS, transposing inside the DMA ----
  // i -> (c = i/64, p = i%64): 64 consecutive floats per channel from global
  // (coalesced), scattered into LDS at stride 85 floats (85 coprime 64 banks
  // -> conflict-free).
  for (int i = tid; i < kCH * kPT; i += kThreads) {
    const int c = i >> 6;                         // i / kPT
    const int p = i & (kPT - 1);                  // i % kPT
    if (p < npos) {
      const unsigned lds_off = (unsigned)(uintptr_t)&tile[p * kCH + c];
      async_f32_to_lds(lds_off, slab + (size_t)c * kGG + (pos0 + p));
    }
  }
  asm volatile("s_wait_asynccnt 0" ::: "memory"); // drain this wave's DMAs
  __syncthreads();                                // all waves' tiles visible

  // Anchor scales; (ANCHOR/ratio)*ratio folds bit-exactly to ANCHOR.
  const float ax = (a == 0) ? 116.0f : (a == 1) ? 156.0f : 373.0f;
  const float ay = (a == 0) ?  90.0f : (a == 1) ? 198.0f : 326.0f;

  // ---- Decode + vectorized coalesced NT stores ----
  // npos*85 is a multiple of 4 (5440 or 1360) and pos0*85 is a multiple of 4,
  // so v4f access is 16B-aligned on both LDS and global sides.
  const int total4 = (npos * kCH) >> 2;
  float* const otile = oslab + (size_t)pos0 * kCH;

  for (int i4 = tid; i4 < total4; i4 += kThreads) {
    const int i = i4 << 2;
    const v4f v = *reinterpret_cast<const v4f*>(&tile[i]);  // ds_load_b128
    v4f o;
#pragma unroll
    for (int j = 0; j < 4; ++j) {
      const int idx = i + j;                      // idx = p*85 + c
      const int p   = idx / kCH;
      const int c   = idx - p * kCH;
      const float x = v[j];
      float r;
      if (c >= 5) {
        r = x;                                    // class logits: raw copy
      } else if (c == 0) {
        const int pos = pos0 + p;
        const int gx  = pos - (pos / kG) * kG;
        r = (sigmoidf_(x) + (float)gx) * kRatio;  // bx * ratio
      } else if (c == 1) {
        const int gy = (pos0 + p) / kG;
        r = (sigmoidf_(x) + (float)gy) * kRatio;  // by * ratio
      } else if (c == 2) {
        r = fminf(expf(x), kExpClamp) * ax;       // bw * ratio
      } else if (c == 3) {
        r = fminf(expf(x), kExpClamp) * ay;       // bh * ratio
      } else {                                    // c == 4
        r = sigmoidf_(x);                         // objectness
      }
      o[j] = r;
    }
    __builtin_nontemporal_store(o, reinterpret_cast<v4f*>(&otile[i]));
  }
}

extern "C" void kernel_launch(void* const* d_in, const int* in_sizes, int n_in,
                              void* d_out, int out_size, void* d_ws, size_t ws_size,
                              hipStream_t stream) {
  const float* X = (const float*)d_in[0];
  float* out = (float*)d_out;

  // Derive batch from input size: in_sizes[0] = B * 255 * 52 * 52.
  const int B = in_sizes[0] / (kA * kCH * kGG);

  const dim3 grid(B * kA * kTiles);               // 8256 blocks for B=64
  yolo_head_kernel<<<grid, kThreads, 0, stream>>>(X, out);
}